// DisplaceChannel_1400159338611
// MI455X (gfx1250) — compile-verified
//
#include <hip/hip_runtime.h>

// DisplaceChannel: out[b, p*32+cc, y, x] = inp[b, p*32+cc, y-off_y[p], x-off_x[p]]
// if source coords in bounds, else 0.  Shifts are multiples of 32 -> float4-aligned.

#define HEIGHT 64
#define WIDTH 64
#define NUM_POS 9
#define CHAN_PER_POS 32
#define CHANNELS (NUM_POS * CHAN_PER_POS)   // 288
#define BATCH 16

typedef float vfloat4 __attribute__((ext_vector_type(4)));

__global__ __launch_bounds__(256) void displace_channel_kernel(
    const float* __restrict__ inp,
    const int*   __restrict__ offsets,   // (9,2) flat: [off_x, off_y] per position
    float*       __restrict__ out)
{
    const int tid = threadIdx.x;
    const int x4  = tid & 15;                 // which float4 in the row (16 per row)
    const int yl  = tid >> 4;                 // 0..15 rows handled per block
    const int y   = ((int)blockIdx.x << 4) + yl;
    const int c   = (int)blockIdx.y;          // 0..287
    const int b   = (int)blockIdx.z;          // 0..15
    const int p   = c >> 5;                   // channel group -> grid position

    // p is uniform across the block -> these become scalar (s_load) fetches.
    const int off_x = offsets[2 * p + 0];
    const int off_y = offsets[2 * p + 1];

    const int sy = y - off_y;
    const int sx = (x4 << 2) - off_x;         // stays 16B aligned (shifts are ±32/0)

    const int plane = (b * CHANNELS + c) * (HEIGHT * WIDTH);
    const int x     = x4 << 2;

    vfloat4 v = {0.f, 0.f, 0.f, 0.f};
    // Whole float4 is valid or invalid together since |off_x| in {0,32} >= 4.
    if ((unsigned)sy < (unsigned)HEIGHT && (unsigned)sx <= (unsigned)(WIDTH - 4)) {
        const vfloat4* src =
            (const vfloat4*)(inp + plane + sy * WIDTH + sx);
        v = __builtin_nontemporal_load(src);  // read-once stream: TH=NT
    }
    vfloat4* dst = (vfloat4*)(out + plane + y * WIDTH + x);
    __builtin_nontemporal_store(v, dst);      // write-once stream: TH=NT
}

extern "C" void kernel_launch(void* const* d_in, const int* in_sizes, int n_in,
                              void* d_out, int out_size, void* d_ws, size_t ws_size,
                              hipStream_t stream)
{
    const float* inp     = (const float*)d_in[0];   // (16, 288, 64, 64) f32
    const int*   offsets = (const int*)d_in[1];     // (9, 2) i32
    float*       out     = (float*)d_out;           // (16, 288, 64, 64) f32

    dim3 grid(HEIGHT / 16, CHANNELS, BATCH);        // (4, 288, 16)
    displace_channel_kernel<<<grid, 256, 0, stream>>>(inp, offsets, out);
}